// CrossAttentionLayer_convk3s2_54735063220674
// MI455X (gfx1250) — compile-verified
//
#include <hip/hip_runtime.h>
#include <math.h>

typedef float v2f __attribute__((ext_vector_type(2)));
typedef float v8f __attribute__((ext_vector_type(8)));

#define DD   128
#define NH   8
#define HDIM 16
#define SCALE 0.25f            // (128/8)^-0.5
#define H0   12
#define W0   18
#define L0   (H0*W0)           // 216
#define HC   6
#define WC   9
#define LC   (HC*WC)           // 54
#define TOKS 132               // padded LDS row stride (floats)
#define TCS  132

// workspace layout (floats)
#define WS_UF 0                // u_fine  [6][8][128]
#define WS_OF 6144             // off_fine[6][8]
#define WS_UC 6208             // u_coarse[4][8][128]
#define WS_OC 10304            // off_coarse[4][8]

// dynamic LDS layout (floats)
#define SM_TOK   0             // 216*132 = 28512
#define SM_TC    28512         // 54*132  = 7128
#define SM_XATTN 35640         // 10*128  (also reused as MLP hidden)
#define SM_XROW  36920         // 10*128
#define SM_XN    38200         // 10*128
#define SM_WSC   39480         // 8 waves * 304 -> ends 41912
#define SM_ZROW  41912         // 132-float zero row (conv OOB fragments)
#define SMEM_FLOATS 42044

__device__ __forceinline__ float wredmax(float v) {
  #pragma unroll
  for (int m = 16; m >= 1; m >>= 1) v = fmaxf(v, __shfl_xor(v, m, 32));
  return v;
}
__device__ __forceinline__ float wredsum(float v) {
  #pragma unroll
  for (int m = 16; m >= 1; m >>= 1) v += __shfl_xor(v, m, 32);
  return v;
}

// ---------------------------------------------------------------------------
// Prep: LN(query) -> q projections -> fold q into K-projections:
//   u[{win,head}][c] = SCALE * sum_d wk[c, h*16+d] * q[h*16+d]
//   off[{win,head}]  = SCALE * sum_d bk[h*16+d]    * q[h*16+d]
// ---------------------------------------------------------------------------
__global__ void __launch_bounds__(128)
prep_kernel(const float* __restrict__ query,
            const float* __restrict__ ln1g, const float* __restrict__ ln1b,
            const float* __restrict__ ln1cg, const float* __restrict__ ln1cb,
            const float* __restrict__ wq, const float* __restrict__ bq,
            const float* __restrict__ wqc, const float* __restrict__ bqc,
            const float* __restrict__ wk, const float* __restrict__ bk,
            const float* __restrict__ wkc, const float* __restrict__ bkc,
            float* __restrict__ ws) {
  __shared__ float red[128];
  __shared__ float qn[10 * DD];
  __shared__ float qp[10 * DD];
  const int tid = threadIdx.x;

  for (int r = 0; r < 10; ++r) {
    float x = query[r * DD + tid];
    red[tid] = x; __syncthreads();
    for (int s = 64; s > 0; s >>= 1) { if (tid < s) red[tid] += red[tid + s]; __syncthreads(); }
    float m = red[0] * (1.0f / 128.0f); __syncthreads();
    float d = x - m;
    red[tid] = d * d; __syncthreads();
    for (int s = 64; s > 0; s >>= 1) { if (tid < s) red[tid] += red[tid + s]; __syncthreads(); }
    float var = red[0] * (1.0f / 128.0f); __syncthreads();
    const float* g  = (r < 6) ? ln1g : ln1cg;
    const float* bb = (r < 6) ? ln1b : ln1cb;
    qn[r * DD + tid] = d * (1.0f / sqrtf(var + 1e-5f)) * g[tid] + bb[tid];
  }
  __syncthreads();

  for (int r = 0; r < 10; ++r) {
    const float* w    = (r < 6) ? wq : wqc;
    const float* bias = (r < 6) ? bq : bqc;
    float acc = bias[tid];
    for (int c = 0; c < DD; ++c) acc += qn[r * DD + c] * w[c * DD + tid];
    qp[r * DD + tid] = acc;
  }
  __syncthreads();

  // fine: 6 windows x 8 heads x 128 channels
  for (int idx = tid; idx < 6 * NH * DD; idx += 128) {
    int i = idx >> 10, h = (idx >> 7) & 7, c = idx & 127;
    float s = 0.f;
    #pragma unroll
    for (int d = 0; d < HDIM; ++d) s += wk[c * DD + h * HDIM + d] * qp[i * DD + h * HDIM + d];
    ws[WS_UF + idx] = SCALE * s;
  }
  if (tid < 48) {
    int i = tid >> 3, h = tid & 7;
    float s = 0.f;
    #pragma unroll
    for (int d = 0; d < HDIM; ++d) s += bk[h * HDIM + d] * qp[i * DD + h * HDIM + d];
    ws[WS_OF + tid] = SCALE * s;
  }
  // coarse: 4 windows x 8 heads x 128 channels
  for (int idx = tid; idx < 4 * NH * DD; idx += 128) {
    int i = idx >> 10, h = (idx >> 7) & 7, c = idx & 127;
    float s = 0.f;
    #pragma unroll
    for (int d = 0; d < HDIM; ++d) s += wkc[c * DD + h * HDIM + d] * qp[(6 + i) * DD + h * HDIM + d];
    ws[WS_UC + idx] = SCALE * s;
  }
  if (tid < 32) {
    int i = tid >> 3, h = tid & 7;
    float s = 0.f;
    #pragma unroll
    for (int d = 0; d < HDIM; ++d) s += bkc[h * HDIM + d] * qp[(6 + i) * DD + h * HDIM + d];
    ws[WS_OC + tid] = SCALE * s;
  }
}

// ---------------------------------------------------------------------------
// Fused per-batch-element kernel: one workgroup (8 wave32) per b.
// ---------------------------------------------------------------------------
__global__ void __launch_bounds__(256)
fused_kernel(const float* __restrict__ query,
             const float* __restrict__ tgt,
             const float* __restrict__ wv,  const float* __restrict__ bv,
             const float* __restrict__ wp,  const float* __restrict__ bp,
             const float* __restrict__ f1w, const float* __restrict__ f1b,
             const float* __restrict__ f2w, const float* __restrict__ f2b,
             const float* __restrict__ cwg, const float* __restrict__ cb,
             const float* __restrict__ wvc, const float* __restrict__ bvc,
             const float* __restrict__ wpc, const float* __restrict__ bpc,
             const float* __restrict__ f1cw, const float* __restrict__ f1cb,
             const float* __restrict__ f2cw, const float* __restrict__ f2cb,
             const float* __restrict__ ln2g, const float* __restrict__ ln2b,
             const float* __restrict__ ln2cg, const float* __restrict__ ln2cb,
             const float* __restrict__ ws,
             float* __restrict__ out) {
  extern __shared__ float smem[];
  float* tokL  = smem + SM_TOK;
  float* tcL   = smem + SM_TC;
  float* xattn = smem + SM_XATTN;   // attn outputs; reused as MLP hidden
  float* xrow  = smem + SM_XROW;
  float* xnb   = smem + SM_XN;

  const int b    = blockIdx.x;
  const int tid  = threadIdx.x;
  const int wave = tid >> 5;
  const int lane = tid & 31;
  float* wscr = smem + SM_WSC + wave * 304;   // uloc[128] | a[48] | tbar[128]
  float* uloc = wscr;
  float* aarr = wscr + 128;
  float* tbar = wscr + 176;

  // ---- 1. stream the 216x128 token tile into LDS (read tgt exactly once) --
  {
    const float4* src = (const float4*)(tgt + (size_t)b * (L0 * DD));
    for (int idx = tid; idx < L0 * DD / 4; idx += 256) {
      float4 v = src[idx];
      int row = idx >> 5, col = (idx & 31) << 2;
      *(float4*)&tokL[row * TOKS + col] = v;
    }
    // zero row used by OOB conv A-fragments (branch-free padding)
    for (int idx = tid; idx < TOKS; idx += 256) smem[SM_ZROW + idx] = 0.f;
  }
  __syncthreads();

  // ---- 2. fine attention: 48 (window,head) pairs, 6 per wave -------------
  for (int j = 0; j < 6; ++j) {
    const int pp = wave * 6 + j;
    const int i = pp >> 3, h = pp & 7;
    const int wy0 = (i / 3) * 6, wx0 = (i % 3) * 6;
    const float* u = ws + WS_UF + pp * DD;
    const float offv = ws[WS_OF + pp];
    for (int c = lane; c < DD; c += 32) uloc[c] = u[c];
    // scores: lane owns token k=lane (and k=lane+32 when lane<4); 36 tokens
    int k1 = lane + 32; if (k1 > 35) k1 = 35;
    const int row0 = ((wy0 + lane / 6) * W0 + wx0 + lane % 6) * TOKS;
    const int row1 = ((wy0 + k1 / 6) * W0 + wx0 + k1 % 6) * TOKS;
    float s0 = offv, s1 = offv;
    for (int c = 0; c < DD; ++c) {
      float uu = uloc[c];
      s0 += uu * tokL[row0 + c];
      s1 += uu * tokL[row1 + c];
    }
    float s1m = (lane < 4) ? s1 : -3.0e38f;
    float m = wredmax(fmaxf(s0, s1m));
    float e0 = expf(s0 - m);
    float e1 = (lane < 4) ? expf(s1 - m) : 0.f;
    float inv = 1.0f / wredsum(e0 + e1);
    aarr[lane] = e0 * inv;
    if (lane < 4) aarr[lane + 32] = e1 * inv;
    // softmax-weighted *token* average (V-proj folded past softmax)
    const int c4 = lane * 4;
    float t0 = 0, t1 = 0, t2 = 0, t3 = 0;
    for (int k = 0; k < 36; ++k) {
      float wg = aarr[k];
      int row = ((wy0 + k / 6) * W0 + wx0 + k % 6) * TOKS + c4;
      t0 += wg * tokL[row];     t1 += wg * tokL[row + 1];
      t2 += wg * tokL[row + 2]; t3 += wg * tokL[row + 3];
    }
    tbar[c4] = t0; tbar[c4 + 1] = t1; tbar[c4 + 2] = t2; tbar[c4 + 3] = t3;
    if (lane < 16) {
      int d = h * HDIM + lane;
      float acc = bv[d];
      for (int c = 0; c < DD; ++c) acc += tbar[c] * wv[c * DD + d];
      xattn[i * DD + d] = acc;
    }
  }
  __syncthreads();

  // ---- 3a. fine residual + output projection ------------------------------
  for (int rep = 0; rep < 3; ++rep) {
    int idx = rep * 256 + tid;
    int i = idx >> 7, d = idx & 127;
    float acc = query[i * DD + d] + bp[d];
    for (int c = 0; c < DD; ++c) acc += xattn[i * DD + c] * wp[c * DD + d];
    xrow[i * DD + d] = acc;
  }

  // ---- 3b. conv 3x3/s2 as WMMA GEMM: tc[54,128] = im2col(tok) @ cw -------
  {
    const int nt = wave;               // N tile (16 output channels)
    const int l15 = lane & 15;
    const int khalf = lane >> 4;       // A/B fragment K-half
    int ycv[4], xcv[4]; bool pvv[4];
    #pragma unroll
    for (int mt = 0; mt < 4; ++mt) {
      int p = mt * 16 + l15;
      pvv[mt] = (p < LC);
      int pc = pvv[mt] ? p : 0;
      ycv[mt] = pc / WC; xcv[mt] = pc % WC;
    }
    const float cbv = cb[nt * 16 + l15];
    v8f acc[4];
    #pragma unroll
    for (int mt = 0; mt < 4; ++mt)
      #pragma unroll
      for (int r = 0; r < 8; ++r) acc[mt][r] = cbv;

    for (int tap = 0; tap < 9; ++tap) {
      const int dy = tap / 3, dx = tap % 3;
      int rb[4];
      #pragma unroll
      for (int mt = 0; mt < 4; ++mt) {
        int ty = 2 * ycv[mt] + dy - 1;
        int tx = 2 * xcv[mt] + dx - 1;
        bool ib = pvv[mt] && ty >= 0 && ty < H0 && tx >= 0 && tx < W0;
        // OOB lanes read the shared zero row: branch-free fragment loads
        rb[mt] = ib ? (ty * W0 + tx) * TOKS : SM_ZROW;
      }
      const float* cwt = cwg + tap * DD * DD + nt * 16 + l15;
      for (int k4 = 0; k4 < 32; ++k4) {
        const int ci0 = k4 * 4 + khalf * 2;
        // B fragment from L2-resident conv weights (issued first: latency
        // overlaps the DS loads below)
        v2f bfrag;
        bfrag.x = cwt[ci0 * DD];
        bfrag.y = cwt[(ci0 + 1) * DD];
        // batch all four A fragments (clauseable ds_load_b64, no exec mask)
        v2f af[4];
        #pragma unroll
        for (int mt = 0; mt < 4; ++mt)
          af[mt] = *(const v2f*)&smem[rb[mt] + ci0];
        #pragma unroll
        for (int mt = 0; mt < 4; ++mt)
          acc[mt] = __builtin_amdgcn_wmma_f32_16x16x4_f32(
              false, af[mt], false, bfrag, (short)0, acc[mt], false, false);
      }
    }
    #pragma unroll
    for (int mt = 0; mt < 4; ++mt)
      #pragma unroll
      for (int r = 0; r < 8; ++r) {
        int row = mt * 16 + khalf * 8 + r;
        if (row < LC) tcL[row * TCS + nt * 16 + l15] = acc[mt][r];
      }
  }
  __syncthreads();

  // ---- 4. coarse attention: 32 (window,head) pairs, 4 per wave -----------
  for (int j = 0; j < 4; ++j) {
    const int pp = wave * 4 + j;
    const int i = pp >> 3, h = pp & 7;
    const int hi = i >> 1, wi = i & 1;
    const int wy0 = hi * 3, wx0 = wi * 4;
    const int nx = wi ? 5 : 4;
    const int nc = 3 * nx;             // 12 or 15
    const float* u = ws + WS_UC + pp * DD;
    const float offv = ws[WS_OC + pp];
    for (int c = lane; c < DD; c += 32) uloc[c] = u[c];
    int kc = (lane < nc) ? lane : (nc - 1);
    const int row0 = ((wy0 + kc / nx) * WC + wx0 + kc % nx) * TCS;
    float s0 = offv;
    for (int c = 0; c < DD; ++c) s0 += uloc[c] * tcL[row0 + c];
    float m = wredmax((lane < nc) ? s0 : -3.0e38f);
    float e0 = (lane < nc) ? expf(s0 - m) : 0.f;
    float inv = 1.0f / wredsum(e0);
    aarr[lane] = e0 * inv;
    const int c4 = lane * 4;
    float t0 = 0, t1 = 0, t2 = 0, t3 = 0;
    for (int k = 0; k < nc; ++k) {
      float wg = aarr[k];
      int row = ((wy0 + k / nx) * WC + wx0 + k % nx) * TCS + c4;
      t0 += wg * tcL[row];     t1 += wg * tcL[row + 1];
      t2 += wg * tcL[row + 2]; t3 += wg * tcL[row + 3];
    }
    tbar[c4] = t0; tbar[c4 + 1] = t1; tbar[c4 + 2] = t2; tbar[c4 + 3] = t3;
    if (lane < 16) {
      int d = h * HDIM + lane;
      float acc = bvc[d];
      for (int c = 0; c < DD; ++c) acc += tbar[c] * wvc[c * DD + d];
      xattn[(6 + i) * DD + d] = acc;
    }
  }
  __syncthreads();

  // ---- 5. coarse residual + output projection ----------------------------
  for (int rep = 0; rep < 2; ++rep) {
    int idx = rep * 256 + tid;
    int i = idx >> 7, d = idx & 127;
    float acc = query[(6 + i) * DD + d] + bpc[d];
    for (int c = 0; c < DD; ++c) acc += xattn[(6 + i) * DD + c] * wpc[c * DD + d];
    xrow[(6 + i) * DD + d] = acc;
  }
  __syncthreads();

  // ---- 6. LayerNorm over all 10 rows (wave-per-row) ----------------------
  for (int pass = 0; pass < 2; ++pass) {
    int r = wave + pass * 8;
    if (r < 10) {
      const int c4 = lane * 4;
      float x0 = xrow[r * DD + c4],     x1 = xrow[r * DD + c4 + 1];
      float x2 = xrow[r * DD + c4 + 2], x3 = xrow[r * DD + c4 + 3];
      float mean = wredsum(x0 + x1 + x2 + x3) * (1.0f / 128.0f);
      float d0 = x0 - mean, d1 = x1 - mean, d2 = x2 - mean, d3 = x3 - mean;
      float var = wredsum(d0 * d0 + d1 * d1 + d2 * d2 + d3 * d3) * (1.0f / 128.0f);
      float inv = 1.0f / sqrtf(var + 1e-5f);
      const float* g  = (r < 6) ? ln2g : ln2cg;
      const float* bb = (r < 6) ? ln2b : ln2cb;
      xnb[r * DD + c4]     = d0 * inv * g[c4]     + bb[c4];
      xnb[r * DD + c4 + 1] = d1 * inv * g[c4 + 1] + bb[c4 + 1];
      xnb[r * DD + c4 + 2] = d2 * inv * g[c4 + 2] + bb[c4 + 2];
      xnb[r * DD + c4 + 3] = d3 * inv * g[c4 + 3] + bb[c4 + 3];
    }
  }
  __syncthreads();

  // ---- 7. MLP fc1 + exact GELU (hidden overwrites xattn) -----------------
  for (int rep = 0; rep < 5; ++rep) {
    int idx = rep * 256 + tid;
    int r = idx >> 7, d = idx & 127;
    const float* w1 = (r < 6) ? f1w : f1cw;
    const float* b1 = (r < 6) ? f1b : f1cb;
    float acc = b1[d];
    for (int c = 0; c < DD; ++c) acc += xnb[r * DD + c] * w1[c * DD + d];
    xattn[r * DD + d] = 0.5f * acc * (1.0f + erff(acc * 0.70710678118654752f));
  }
  __syncthreads();

  // ---- 8. MLP fc2 + residual -> global output ----------------------------
  for (int rep = 0; rep < 5; ++rep) {
    int idx = rep * 256 + tid;
    int r = idx >> 7, d = idx & 127;
    const float* w2 = (r < 6) ? f2w : f2cw;
    const float* b2 = (r < 6) ? f2b : f2cb;
    float acc = xrow[r * DD + d] + b2[d];
    for (int c = 0; c < DD; ++c) acc += xattn[r * DD + c] * w2[c * DD + d];
    out[(size_t)b * (10 * DD) + idx] = acc;
  }
}

extern "C" void kernel_launch(void* const* d_in, const int* in_sizes, int n_in,
                              void* d_out, int out_size, void* d_ws, size_t ws_size,
                              hipStream_t stream) {
  const float* query = (const float*)d_in[0];
  const float* tgt   = (const float*)d_in[1];
  const float* ln1g  = (const float*)d_in[2];
  const float* ln1b  = (const float*)d_in[3];
  const float* ln2g  = (const float*)d_in[4];
  const float* ln2b  = (const float*)d_in[5];
  const float* wq    = (const float*)d_in[6];
  const float* bq    = (const float*)d_in[7];
  const float* wk    = (const float*)d_in[8];
  const float* bk    = (const float*)d_in[9];
  const float* wv    = (const float*)d_in[10];
  const float* bv    = (const float*)d_in[11];
  const float* wp    = (const float*)d_in[12];
  const float* bp    = (const float*)d_in[13];
  const float* f1w   = (const float*)d_in[14];
  const float* f1b   = (const float*)d_in[15];
  const float* f2w   = (const float*)d_in[16];
  const float* f2b   = (const float*)d_in[17];
  const float* cw    = (const float*)d_in[18];
  const float* cb    = (const float*)d_in[19];
  const float* ln1cg = (const float*)d_in[20];
  const float* ln1cb = (const float*)d_in[21];
  const float* ln2cg = (const float*)d_in[22];
  const float* ln2cb = (const float*)d_in[23];
  const float* wqc   = (const float*)d_in[24];
  const float* bqc   = (const float*)d_in[25];
  const float* wkc   = (const float*)d_in[26];
  const float* bkc   = (const float*)d_in[27];
  const float* wvc   = (const float*)d_in[28];
  const float* bvc   = (const float*)d_in[29];
  const float* wpc   = (const float*)d_in[30];
  const float* bpc   = (const float*)d_in[31];
  const float* f1cw  = (const float*)d_in[32];
  const float* f1cb  = (const float*)d_in[33];
  const float* f2cw  = (const float*)d_in[34];
  const float* f2cb  = (const float*)d_in[35];
  // d_in[36], d_in[37] are H, W (reference fixes them to 12, 18)

  float* ws   = (float*)d_ws;
  float* outp = (float*)d_out;
  const int Bn = in_sizes[1] / (L0 * DD);

  prep_kernel<<<1, 128, 0, stream>>>(query, ln1g, ln1b, ln1cg, ln1cb,
                                     wq, bq, wqc, bqc, wk, bk, wkc, bkc, ws);

  const size_t shmem = (size_t)SMEM_FLOATS * sizeof(float);
  fused_kernel<<<Bn, 256, shmem, stream>>>(
      query, tgt, wv, bv, wp, bp, f1w, f1b, f2w, f2b, cw, cb,
      wvc, bvc, wpc, bpc, f1cw, f1cb, f2cw, f2cb,
      ln2g, ln2b, ln2cg, ln2cb, ws, outp);
}